// PosePostProcessor_3796751089752
// MI455X (gfx1250) — compile-verified
//
#include <hip/hip_runtime.h>
#include <hip/hip_bf16.h>
#include <stdint.h>

// Pose post-processing for MI455X (gfx1250, wave32).
// 544 channels of 256x256 fp32. Memory-bound (~142.6 MB @ 23.3 TB/s ~ 6us floor).
// CDNA5 paths: async global->LDS tile staging (global_load_async_to_lds_b128 +
// s_wait_asynccnt) feeding an LDS-resident 3x3 NMS stencil; wave32 ballot to
// build a candidate bitmap with zero atomics. No WMMA: NMS/top-k are order
// statistics with no matmul structure.

#define NUM_KP   17
#define BATCH    32
#define NCH      (BATCH * NUM_KP)   // 544
#define HH       256
#define WW       256
#define NPIX     (HH * WW)          // 65536
#define NWORDS   (NPIX / 32)        // 2048 bitmap words
#define MAXP     30
#define THRESH   0.1f
#define NBINS    1024
#define CAP      1024
#define TILE     32
#define TROWS    (TILE + 2)         // 34 rows incl. halo
#define BLOCK    256

#define NEG_FILL (-1.0e30f)
#define CONSUMED (-3.0e38f)

// Low 32 bits of a generic pointer to LDS == byte offset within the wave's LDS
// allocation (what the async-load VDST VGPR expects).
__device__ __forceinline__ unsigned lds_offset_of(const void* p) {
  return (unsigned)(unsigned long long)p;
}

// Async DMA of 16 bytes per active lane from global memory into LDS.
// GV addressing: vdst = LDS byte offset, vaddr = 64-bit global address.
__device__ __forceinline__ void async_load_b128(unsigned lds_off, const void* gaddr) {
  asm volatile("global_load_async_to_lds_b128 %0, %1, off"
               :: "v"(lds_off), "v"(gaddr)
               : "memory");
}

__device__ __forceinline__ void wait_async0() {
#if defined(__has_builtin) && __has_builtin(__builtin_amdgcn_s_wait_asynccnt)
  __builtin_amdgcn_s_wait_asynccnt(0);
#else
  asm volatile("s_wait_asynccnt 0x0" ::: "memory");
#endif
}

__device__ __forceinline__ unsigned ballot32(bool p) {
#if defined(__has_builtin) && __has_builtin(__builtin_amdgcn_ballot_w32)
  return __builtin_amdgcn_ballot_w32(p);
#else
  return (unsigned)__ballot((int)p);
#endif
}

// Bit-pattern bins: monotonic for positive floats.
__device__ __forceinline__ unsigned coarse_bin(float v) {
  return __float_as_uint(v) >> 21;       // [0, 1024) for positive floats
}
__device__ __forceinline__ unsigned fine_bin(float v) {
  return (__float_as_uint(v) >> 11) & 1023u;
}

__global__ __launch_bounds__(BLOCK)
void pose_topk_kernel(const float* __restrict__ hm,
                      int* __restrict__ coords,
                      float* __restrict__ scores,
                      unsigned char* __restrict__ valid)
{
  __shared__ __align__(16) float stile[TROWS * WW];   // 34.0 KB tile (+halo)
  __shared__ unsigned hist[NBINS];                    //  4 KB
  __shared__ unsigned cbm[NWORDS];                    //  8 KB candidate bitmap
  __shared__ float listV[CAP];                        //  4 KB
  __shared__ int   listI[CAP];                        //  4 KB
  __shared__ float redV[BLOCK];
  __shared__ int   redI[BLOCK];
  __shared__ int   redP[BLOCK];
  __shared__ int sB1, sF1, sNeed, sTakeAll, sCnt, sPadCur;

  const int tid = threadIdx.x;
  const int ch  = blockIdx.x;
  const float* chp = hm + (size_t)ch * NPIX;

  for (int i = tid; i < NBINS; i += BLOCK) hist[i] = 0u;
  if (tid == 0) sCnt = 0;
  __syncthreads();

  // --- Phase A: async-staged tiles -> rolling-window NMS -> coarse hist + bitmap
  // Each thread owns column cc == tid; lanes of a wave are 32 consecutive
  // columns, so ballot() yields one bitmap word per (row, 32-col group).
  const int cc = tid;
  const int c0 = (cc > 0)   ? cc - 1 : cc;
  const int c1 = (cc < 255) ? cc + 1 : cc;
  const int lane     = tid & 31;
  const int wordBase = tid >> 5;       // which 32-col group this wave covers

  for (int ti = 0; ti < HH / TILE; ++ti) {
    const int r0  = ti * TILE;
    const int tlo = (r0 == 0) ? 1 : 0;               // first staged tile row
    const int thi = (r0 + TILE >= HH) ? (TROWS - 2) : (TROWS - 1);

    if (tlo == 1)          stile[tid]                    = NEG_FILL; // above image
    if (thi == TROWS - 2)  stile[(TROWS - 1) * WW + tid] = NEG_FILL; // below image

    const int   nbytes = (thi - tlo + 1) * WW * 4;
    const char* gsrc   = (const char*)(chp + (r0 - 1 + tlo) * WW);
    const unsigned lbase = lds_offset_of(&stile[tlo * WW]);
    for (int off = tid * 16; off < nbytes; off += BLOCK * 16)
      async_load_b128(lbase + (unsigned)off, gsrc + off);   // 16B/lane DMA
    wait_async0();
    __syncthreads();

    // Rolling 3-row window in registers: 3 LDS loads per output pixel.
    float a0 = stile[0 * WW + c0], a1 = stile[0 * WW + cc], a2 = stile[0 * WW + c1];
    float b0 = stile[1 * WW + c0], b1 = stile[1 * WW + cc], b2 = stile[1 * WW + c1];
    for (int rr = 0; rr < TILE; ++rr) {
      const float d0 = stile[(rr + 2) * WW + c0];
      const float d1 = stile[(rr + 2) * WW + cc];
      const float d2 = stile[(rr + 2) * WW + c1];
      const float v  = b1;
      float m = fmaxf(fmaxf(fmaxf(a0, a1), fmaxf(a2, b0)),
                      fmaxf(fmaxf(b2, d0), fmaxf(d1, d2)));
      const bool cand = (v > THRESH) && (v >= m);   // ties kept (hmax==v)
      const unsigned word = ballot32(cand);         // convergent: no divergence above
      if (lane == 0) cbm[(r0 + rr) * (WW / 32) + wordBase] = word;
      if (cand) atomicAdd(&hist[coarse_bin(v)], 1u);
      a0 = b0; a1 = b1; a2 = b2;
      b0 = d0; b1 = d1; b2 = d2;
    }
    __syncthreads();   // stile reuse + bitmap visibility
  }

  // --- Phase B: coarse threshold bin -----------------------------------------
  if (tid == 0) {
    int run = 0, b1c = -1, above = 0;
    for (int b = NBINS - 1; b >= 0; --b) {
      run += (int)hist[b];
      if (run >= MAXP) { b1c = b; above = run - (int)hist[b]; break; }
    }
    if (b1c < 0) { sTakeAll = 1; sB1 = 0; sF1 = 0; sNeed = 0; }
    else         { sTakeAll = 0; sB1 = b1c; sNeed = MAXP - above; }
  }
  __syncthreads();
  const int b1c     = sB1;
  const int takeAll = sTakeAll;

  // --- Phase C: refine within coarse bin b1 (sparse walk over bitmap) --------
  for (int i = tid; i < NBINS; i += BLOCK) hist[i] = 0u;
  __syncthreads();
  if (!takeAll) {
    for (int w = tid; w < NWORDS; w += BLOCK) {
      unsigned bits = cbm[w];
      while (bits) {
        const int b = __ffs(bits) - 1;
        bits &= bits - 1;
        const int   idx = w * 32 + b;
        const float v   = chp[idx];                 // L2-resident re-read
        if ((int)coarse_bin(v) == b1c) atomicAdd(&hist[fine_bin(v)], 1u);
      }
    }
  }
  __syncthreads();
  if (tid == 0 && !takeAll) {
    int run = 0, f1c = 0;
    for (int f = NBINS - 1; f >= 0; --f) {
      run += (int)hist[f];
      if (run >= sNeed) { f1c = f; break; }
    }
    sF1 = f1c;
  }
  __syncthreads();
  const int f1c = sF1;

  // --- Phase E: collect candidates above refined threshold -------------------
  for (int w = tid; w < NWORDS; w += BLOCK) {
    unsigned bits = cbm[w];
    while (bits) {
      const int b = __ffs(bits) - 1;
      bits &= bits - 1;
      const int   idx = w * 32 + b;
      const float v   = chp[idx];
      const int   cb  = (int)coarse_bin(v);
      if (cb > b1c || (cb == b1c && (int)fine_bin(v) >= f1c)) {
        const int slot = atomicAdd(&sCnt, 1);
        if (slot < CAP) { listV[slot] = v; listI[slot] = idx; }
      }
    }
  }
  if (tid == 0) sPadCur = 0;
  __syncthreads();
  const int M = (sCnt < CAP) ? sCnt : CAP;

  // --- Phase F: 30 rounds of block argmax (tie-break: lower flat index) ------
  for (int k = 0; k < MAXP; ++k) {
    float bv = CONSUMED; int bi = 0x7FFFFFFF; int bp = -1;
    for (int j = tid; j < M; j += BLOCK) {
      const float v = listV[j];
      const int   i = listI[j];
      if (v > bv || (v == bv && i < bi)) { bv = v; bi = i; bp = j; }
    }
    redV[tid] = bv; redI[tid] = bi; redP[tid] = bp;
    __syncthreads();
    for (int s = BLOCK / 2; s > 0; s >>= 1) {
      if (tid < s) {
        const float ov = redV[tid + s]; const int oi = redI[tid + s];
        if (ov > redV[tid] || (ov == redV[tid] && oi < redI[tid])) {
          redV[tid] = ov; redI[tid] = oi; redP[tid] = redP[tid + s];
        }
      }
      __syncthreads();
    }
    if (tid == 0) {
      const size_t base = (size_t)ch * MAXP + k;
      if (redP[0] >= 0 && redV[0] > -1.0e29f) {
        const int idx = redI[0];
        coords[base * 2 + 0] = (idx & 255) * 4;   // x * STRIDE
        coords[base * 2 + 1] = (idx >> 8) * 4;    // y * STRIDE
        scores[base] = redV[0];
        valid[base]  = 1;
        listV[redP[0]] = CONSUMED;                // consume for later rounds
      } else {
        // Padding: mimic top_k over the -inf-masked tail -> lowest unused
        // non-candidate flat indices, score = -inf, valid = false.
        int p = sPadCur;
        for (;;) {
          bool hit = false;
          for (int j = 0; j < M; ++j) if (listI[j] == p) { hit = true; break; }
          if (!hit) break;
          ++p;
        }
        coords[base * 2 + 0] = (p & 255) * 4;
        coords[base * 2 + 1] = (p >> 8) * 4;
        scores[base] = -__builtin_huge_valf();
        valid[base]  = 0;
        sPadCur = p + 1;
      }
    }
    __syncthreads();
  }
}

extern "C" void kernel_launch(void* const* d_in, const int* in_sizes, int n_in,
                              void* d_out, int out_size, void* d_ws, size_t ws_size,
                              hipStream_t stream) {
  (void)in_sizes; (void)n_in; (void)out_size; (void)d_ws; (void)ws_size;
  const float* hm = (const float*)d_in[0];

  // Tuple output, concatenated flat in return order:
  //   coords  int32  [32,17,30,2]  -> 32640 * 4 B
  //   scores  f32    [32,17,30]    -> 16320 * 4 B
  //   valid   bool   [32,17,30]    -> 16320 * 1 B
  char* out = (char*)d_out;
  int*           coords = (int*)out;
  float*         scores = (float*)(out + (size_t)NCH * MAXP * 2 * sizeof(int));
  unsigned char* valid  = (unsigned char*)(out + (size_t)NCH * MAXP * 2 * sizeof(int)
                                               + (size_t)NCH * MAXP * sizeof(float));

  pose_topk_kernel<<<dim3(NCH), dim3(BLOCK), 0, stream>>>(hm, coords, scores, valid);
}